// PatchMerging_8985071583247
// MI455X (gfx1250) — compile-verified
//
#include <hip/hip_runtime.h>

typedef __attribute__((ext_vector_type(2))) float v2f;
typedef __attribute__((ext_vector_type(8))) float v8f;

// D = A(16x4,f32) * B(4x16,f32) + C(16x16,f32)
#define WMMA_F32X4(a, b, c) \
    __builtin_amdgcn_wmma_f32_16x16x4_f32(false, (a), false, (b), (short)0, (c), false, false)

// problem constants
constexpr int HH = 128, WW = 128;   // feature grid after patchify
constexpr int NHD = 4;              // heads
constexpr int STA = 17;             // LDS row stride (16 + 1 pad)
constexpr int STB = 65;             // LDS row stride (64 + 1 pad)

// ---------------------------------------------------------------------------
// Kernel 1: fused patchify + 4x4 window attention + proj.
// One wave32 per window (16 tokens x 16 features). 8 waves per workgroup.
// Output: xr in (B, 128, 128, 16) fp32 layout (workspace).
// ---------------------------------------------------------------------------
__global__ __launch_bounds__(256) void win_attn_kernel(
    const float* __restrict__ x,      // (64, 65536, 4)
    const float* __restrict__ qkv_w,  // (16, 48)
    const float* __restrict__ qkv_b,  // (48)
    const float* __restrict__ rpb,    // (49, 4)
    const float* __restrict__ proj_w, // (16, 16)
    const float* __restrict__ proj_b, // (16)
    float* __restrict__ xr)           // (64, 128, 128, 16)
{
    __shared__ float sm[8 * 6 * 16 * STA];

    const int lane = threadIdx.x & 31;
    const int wv   = threadIdx.x >> 5;
    const int widx = blockIdx.x * 8 + wv;          // window id, < 65536
    const int b  = widx >> 10;                     // batch
    const int wh = (widx >> 5) & 31;               // window row
    const int wc = widx & 31;                      // window col

    float* X = sm + wv * (6 * 16 * STA);
    float* Q = X + 16 * STA;
    float* K = Q + 16 * STA;
    float* V = K + 16 * STA;
    float* S = V + 16 * STA;
    float* O = S + 16 * STA;

    const int ln = lane & 15;                      // N index / row index
    const int hf = lane >> 4;                      // half-wave select

    // ---- gather window tile: token t=(hh,ww), feature f=a*4+c ----
    for (int idx = lane; idx < 256; idx += 32) {
        int t = idx >> 4, f = idx & 15;
        int hh = t >> 2, ww = t & 3;
        int gh = wh * 4 + hh, gw = wc * 4 + ww;
        int a = f >> 2, c = f & 3;
        long long xi = (((long long)b * 65536) + (long long)(gh * WW + gw) * 4 + a) * 4 + c;
        X[t * STA + f] = x[xi];
    }

    // ---- QKV = X @ qkv_w + qkv_b (three 16x16x16 GEMMs, 4 WMMAs each) ----
    {
        v8f dq = {}, dk = {}, dv = {};
        #pragma unroll
        for (int c4 = 0; c4 < 4; ++c4) {
            v2f a;
            a.x = X[ln * STA + c4 * 4 + 2 * hf];
            a.y = X[ln * STA + c4 * 4 + 2 * hf + 1];
            int r0 = (c4 * 4 + 2 * hf) * 48, r1 = (c4 * 4 + 2 * hf + 1) * 48;
            v2f bq; bq.x = qkv_w[r0 + ln];      bq.y = qkv_w[r1 + ln];
            v2f bk; bk.x = qkv_w[r0 + 16 + ln]; bk.y = qkv_w[r1 + 16 + ln];
            v2f bv; bv.x = qkv_w[r0 + 32 + ln]; bv.y = qkv_w[r1 + 32 + ln];
            dq = WMMA_F32X4(a, bq, dq);
            dk = WMMA_F32X4(a, bk, dk);
            dv = WMMA_F32X4(a, bv, dv);
        }
        float bq = qkv_b[ln], bk = qkv_b[16 + ln], bv = qkv_b[32 + ln];
        #pragma unroll
        for (int v = 0; v < 8; ++v) {
            Q[(v + 8 * hf) * STA + ln] = dq[v] + bq;
            K[(v + 8 * hf) * STA + ln] = dk[v] + bk;
            V[(v + 8 * hf) * STA + ln] = dv[v] + bv;
        }
    }

    // ---- per-head attention, accumulate all heads into one 16x16 O ----
    v8f o = {};
    for (int h = 0; h < NHD; ++h) {
        // scores: q_h (16x4) @ k_h^T (4x16), single K=4 WMMA
        v2f a, bb;
        a.x  = Q[ln * STA + 4 * h + 2 * hf];
        a.y  = Q[ln * STA + 4 * h + 2 * hf + 1];
        bb.x = K[ln * STA + 4 * h + 2 * hf];
        bb.y = K[ln * STA + 4 * h + 2 * hf + 1];
        v8f s = {};
        s = WMMA_F32X4(a, bb, s);

        // + relative position bias (closed form), softmax over keys (16 lanes)
        float sv[8];
        #pragma unroll
        for (int v = 0; v < 8; ++v) {
            int p = v + 8 * hf;                         // query token
            int rpi = ((p & 3) - (ln & 3) + 3) * 7 + ((p >> 2) - (ln >> 2) + 3);
            sv[v] = s[v] + rpb[rpi * NHD + h];
        }
        #pragma unroll
        for (int v = 0; v < 8; ++v) {
            float m = sv[v];
            m = fmaxf(m, __shfl_xor(m, 8, 32));
            m = fmaxf(m, __shfl_xor(m, 4, 32));
            m = fmaxf(m, __shfl_xor(m, 2, 32));
            m = fmaxf(m, __shfl_xor(m, 1, 32));
            float e = __expf(sv[v] - m);
            float su = e;
            su += __shfl_xor(su, 8, 32);
            su += __shfl_xor(su, 4, 32);
            su += __shfl_xor(su, 2, 32);
            su += __shfl_xor(su, 1, 32);
            sv[v] = e / su;
        }
        #pragma unroll
        for (int v = 0; v < 8; ++v) S[(v + 8 * hf) * STA + ln] = sv[v];

        // o[:, 4h..4h+3] += S_h @ V_h : head-masked packed-B, K=16 as 4 chunks
        bool mine = ((ln >> 2) == h);
        #pragma unroll
        for (int c4 = 0; c4 < 4; ++c4) {
            v2f as, bv;
            as.x = S[ln * STA + 4 * c4 + 2 * hf];
            as.y = S[ln * STA + 4 * c4 + 2 * hf + 1];
            bv.x = mine ? V[(4 * c4 + 2 * hf) * STA + ln] : 0.0f;
            bv.y = mine ? V[(4 * c4 + 2 * hf + 1) * STA + ln] : 0.0f;
            o = WMMA_F32X4(as, bv, o);
        }
    }

    // ---- proj: O @ proj_w + proj_b ----
    #pragma unroll
    for (int v = 0; v < 8; ++v) O[(v + 8 * hf) * STA + ln] = o[v];
    v8f dp = {};
    #pragma unroll
    for (int c4 = 0; c4 < 4; ++c4) {
        v2f a, bp;
        a.x  = O[ln * STA + 4 * c4 + 2 * hf];
        a.y  = O[ln * STA + 4 * c4 + 2 * hf + 1];
        bp.x = proj_w[(4 * c4 + 2 * hf) * 16 + ln];
        bp.y = proj_w[(4 * c4 + 2 * hf + 1) * 16 + ln];
        dp = WMMA_F32X4(a, bp, dp);
    }
    float pb = proj_b[ln];

    // ---- coalesced store to (B,128,128,16) intermediate ----
    #pragma unroll
    for (int v = 0; v < 8; ++v) {
        int t = v + 8 * hf;
        int hh = t >> 2, ww = t & 3;
        int gh = wh * 4 + hh, gw = wc * 4 + ww;
        xr[(((long long)b * HH + gh) * WW + gw) * 16 + ln] = dp[v] + pb;
    }
}

// ---------------------------------------------------------------------------
// Kernel 2: 2x2 patch-merge gather + LayerNorm(64) + 64x16 reduction GEMM +
// interleaved scatter to (B,256,256). One wave handles 16 output positions.
// ---------------------------------------------------------------------------
__global__ __launch_bounds__(256) void merge_kernel(
    const float* __restrict__ xr,     // (64, 128, 128, 16)
    const float* __restrict__ ln_g,   // (64)
    const float* __restrict__ ln_b,   // (64)
    const float* __restrict__ red_w,  // (64, 16)
    float* __restrict__ out)          // (64, 256, 256)
{
    __shared__ float sm[8 * 16 * STB];
    const int lane = threadIdx.x & 31;
    const int wv   = threadIdx.x >> 5;
    float* T = sm + wv * 16 * STB;
    const int ln = lane & 15;
    const int hf = lane >> 4;
    const long long pbase = ((long long)blockIdx.x * 8 + wv) * 16;   // 16 positions

    // ---- gather 16 x 64 merged tile ----
    for (int idx = lane; idx < 16 * 64; idx += 32) {
        int r = idx >> 6, f64 = idx & 63;
        long long p = pbase + r;
        int b  = (int)(p >> 12);
        int ij = (int)(p & 4095);
        int i = ij >> 6, j = ij & 63;
        int g = f64 >> 4, fi = f64 & 15;
        int dy = g & 1, dx = g >> 1;                  // x0,x1,x2,x3 order
        int gh = 2 * i + dy, gw = 2 * j + dx;
        T[r * STB + f64] = xr[(((long long)b * HH + gh) * WW + gw) * 16 + fi];
    }

    // ---- LayerNorm over 64 channels: two lanes cooperate per row ----
    {
        int r = ln, base = hf * 32;
        float s = 0.0f, s2 = 0.0f;
        #pragma unroll
        for (int f = 0; f < 32; ++f) {
            float v_ = T[r * STB + base + f];
            s += v_; s2 += v_ * v_;
        }
        s  += __shfl_xor(s, 16, 32);
        s2 += __shfl_xor(s2, 16, 32);
        float mu   = s * (1.0f / 64.0f);
        float var  = s2 * (1.0f / 64.0f) - mu * mu;
        float rstd = rsqrtf(var + 1e-5f);
        #pragma unroll
        for (int f = 0; f < 32; ++f) {
            int ff = base + f;
            T[r * STB + ff] = (T[r * STB + ff] - mu) * rstd * ln_g[ff] + ln_b[ff];
        }
    }

    // ---- reduction GEMM: (16x64) @ (64x16), 16 chained f32 WMMAs ----
    v8f d = {};
    #pragma unroll
    for (int c4 = 0; c4 < 16; ++c4) {
        v2f a, bw;
        a.x  = T[ln * STB + 4 * c4 + 2 * hf];
        a.y  = T[ln * STB + 4 * c4 + 2 * hf + 1];
        bw.x = red_w[(4 * c4 + 2 * hf) * 16 + ln];
        bw.y = red_w[(4 * c4 + 2 * hf + 1) * 16 + ln];
        d = WMMA_F32X4(a, bw, d);
    }

    // ---- scatter: out[b, i*4 + f/4, j*4 + f%4] = d ----
    #pragma unroll
    for (int v = 0; v < 8; ++v) {
        long long p = pbase + v + 8 * hf;
        int b  = (int)(p >> 12);
        int ij = (int)(p & 4095);
        int i = ij >> 6, j = ij & 63;
        int c1 = ln >> 2, c2 = ln & 3;
        out[((long long)b * 256 + i * 4 + c1) * 256 + j * 4 + c2] = d[v];
    }
}

// ---------------------------------------------------------------------------
extern "C" void kernel_launch(void* const* d_in, const int* in_sizes, int n_in,
                              void* d_out, int out_size, void* d_ws, size_t ws_size,
                              hipStream_t stream) {
    const float* x      = (const float*)d_in[0];
    const float* qkv_w  = (const float*)d_in[1];
    const float* qkv_b  = (const float*)d_in[2];
    const float* rpb    = (const float*)d_in[3];
    const float* proj_w = (const float*)d_in[4];
    const float* proj_b = (const float*)d_in[5];
    const float* ln_g   = (const float*)d_in[6];
    const float* ln_b   = (const float*)d_in[7];
    const float* red_w  = (const float*)d_in[8];
    float* out = (float*)d_out;
    float* xr  = (float*)d_ws;     // needs 64*128*128*16 floats = 64 MiB

    // 65536 windows / 8 per workgroup
    win_attn_kernel<<<8192, 256, 0, stream>>>(x, qkv_w, qkv_b, rpb, proj_w, proj_b, xr);
    // 262144 positions / (16 per wave * 8 waves)
    merge_kernel<<<2048, 256, 0, stream>>>(xr, ln_g, ln_b, red_w, out);
}